// InfoEnlargeEmbedding_72507637891611
// MI455X (gfx1250) — compile-verified
//
#include <hip/hip_runtime.h>

// Problem constants from the reference: x:[64,1024,256] f32, idxs:[64,2] i32,
// out:[64,1024,768] f32 where out = concat(x, gathered-rows broadcast over L).
#define B_      64
#define L_      1024
#define D_      256
#define K_      2
#define OD_     (D_ * (1 + K_))   // 768 floats per output row
#define TILE_L  8
#define THREADS 256

typedef float v4f __attribute__((ext_vector_type(4)));
typedef __attribute__((address_space(3))) v4f v4f_as3;

__global__ __launch_bounds__(THREADS)
void info_enlarge_embedding_kernel(const float* __restrict__ x,
                                   const int*   __restrict__ idxs,
                                   float*       __restrict__ out) {
    // 2 KB LDS stage for the K gathered rows of this batch (K_*D_ floats).
    __shared__ v4f lds_g[K_ * D_ / 4];          // 128 x 16B

    const int tiles_per_b = L_ / TILE_L;        // 128
    const int b  = blockIdx.x / tiles_per_b;
    const int l0 = (blockIdx.x % tiles_per_b) * TILE_L;
    const int t  = threadIdx.x;

    // ---- Stage gathered rows into LDS via CDNA5 async global->LDS copy ----
    // 128 lanes each move one 16-byte packet (2 rows x 64 v4f).
    if (t < (K_ * D_ / 4)) {
        const int k = t >> 6;                   // which gathered row (0..K-1)
        const int c = t & 63;                   // v4f column within the row
        const int row = idxs[b * K_ + k];       // uniform per 64-lane group
        const float* src = x + (((size_t)b * L_ + (size_t)row) * D_) + (size_t)c * 4;
        v4f_as3* ldst = (v4f_as3*)&lds_g[t];    // 32-bit LDS byte address in a VGPR
        asm volatile("global_load_async_to_lds_b128 %0, %1, off"
                     :: "v"(ldst), "v"(src)
                     : "memory");
    }
    // Wait for this wave's async transfers to land in LDS, then make them
    // visible to the whole workgroup.
    asm volatile("s_wait_asynccnt 0" ::: "memory");
    __syncthreads();

    const size_t row0 = (size_t)b * L_ + (size_t)l0;
    const float* xin  = x   + row0 * D_;
    float*       orow = out + row0 * OD_;

    // ---- Part 1: stream-copy the x tile (TILE_L rows x 64 v4f = 512 v4f) ----
    // Read-once / write-once: non-temporal both ways, b128 width.
    {
        const int c = t & 63;                   // v4f column, invariant
#pragma unroll
        for (int i = 0; i < (TILE_L * D_ / 4) / THREADS; ++i) {   // 2 iters
            const int r = (t >> 6) + i * (THREADS / 64);
            v4f v = __builtin_nontemporal_load((const v4f*)(xin + (size_t)r * D_) + c);
            __builtin_nontemporal_store(v, (v4f*)(orow + (size_t)r * OD_) + c);
        }
    }

    // ---- Part 2: broadcast gathered rows from LDS into every output row ----
    // TILE_L rows x 128 v4f = 1024 v4f; each thread's LDS element is
    // loop-invariant (c = t & 127), so one ds_load_b128 feeds 4 b128 NT stores.
    {
        const int c = t & 127;                  // v4f column in [0,128)
        const v4f v = lds_g[c];                 // ds_load_b128
#pragma unroll
        for (int i = 0; i < (TILE_L * K_ * D_ / 4) / THREADS; ++i) { // 4 iters
            const int r = (t >> 7) + i * (THREADS / 128);
            __builtin_nontemporal_store(v, (v4f*)(orow + (size_t)r * OD_ + D_) + c);
        }
    }
}

extern "C" void kernel_launch(void* const* d_in, const int* in_sizes, int n_in,
                              void* d_out, int out_size, void* d_ws, size_t ws_size,
                              hipStream_t stream) {
    const float* x   = (const float*)d_in[0];
    const int* idxs  = (const int*)d_in[1];
    float* out       = (float*)d_out;

    const int grid = B_ * (L_ / TILE_L);        // 8192 blocks of 256 threads
    info_enlarge_embedding_kernel<<<grid, THREADS, 0, stream>>>(x, idxs, out);
}